// AdvectionDiffusionReaction2M_45578192945901
// MI455X (gfx1250) — compile-verified
//
#include <hip/hip_runtime.h>

// ---------------------------------------------------------------------------
// AdvectionDiffusionReaction, 2 materials. N=512 grid, 199 explicit steps,
// every step's state is an output slab. Memory-bound stencil: per step we
// stage 34x34 halo tiles into LDS via CDNA5 async global->LDS copies
// (ASYNCcnt path), compute the update from LDS, and stream results out.
// Constant divisions strength-reduced to exact-f32 reciprocal multiplies.
// ---------------------------------------------------------------------------

#define GN 512
#define GMB 256
#define GSTEPS 200
#define TILE 32
#define HALO 34           // TILE + 2
#define LDS_STRIDE 36     // pad to dodge bank conflicts
#define NTHREADS 256

#if __has_builtin(__builtin_amdgcn_global_load_async_to_lds_b32)
#define USE_ASYNC_LDS 1
#else
#define USE_ASYNC_LDS 0
#endif

typedef __attribute__((address_space(1))) int as1_int;
typedef __attribute__((address_space(3))) int as3_int;

__global__ __launch_bounds__(NTHREADS) void adr2m_step(
    const float* __restrict__ Tin,
    float* __restrict__ Tout,
    const float* __restrict__ k1p,
    const float* __restrict__ k2p,
    const float* __restrict__ a1p,
    const float* __restrict__ a2p)
{
    __shared__ float s[HALO * LDS_STRIDE];

    const int tid    = threadIdx.x;
    const int base_r = blockIdx.y * TILE;
    const int base_c = blockIdx.x * TILE;

    // uniform scalars (L2-resident, scalar-loaded)
    const float k1 = *k1p, k2 = *k2p, a1 = *a1p, a2 = *a2p;

    // ---- stage clamped (base-1 .. base+32)^2 halo tile into LDS -----------
    for (int e = tid; e < HALO * HALO; e += NTHREADS) {
        int i  = e / HALO;
        int j  = e - i * HALO;
        int gr = base_r - 1 + i; gr = max(0, min(GN - 1, gr));
        int gc = base_c - 1 + j; gc = max(0, min(GN - 1, gc));
        const float* gp = Tin + gr * GN + gc;
        float*       lp = &s[i * LDS_STRIDE + j];
#if USE_ASYNC_LDS
        // LDS aperture: flat addr [31:0] is the DS byte offset (ISA 10.2).
        unsigned lds_off = (unsigned)(unsigned long long)(void*)lp;
        __builtin_amdgcn_global_load_async_to_lds_b32(
            (as1_int*)(unsigned long long)(const void*)gp,
            (as3_int*)lds_off,
            /*imm offset*/0, /*cpol*/0);
#else
        *lp = *gp;
#endif
    }
#if USE_ASYNC_LDS
#if __has_builtin(__builtin_amdgcn_s_wait_asynccnt)
    __builtin_amdgcn_s_wait_asynccnt(0);
#else
    asm volatile("s_wait_asynccnt 0" ::: "memory");
#endif
#endif
    __syncthreads();

    // ---- compute --------------------------------------------------------
    // 1/dx2 = 511*511 = 261121 and 1/(2dx) = 255.5 are EXACT in f32, so
    // multiplying by them is a faithful strength-reduction of the divides.
    const float inv_dx2 = 261121.0f;   // (N-1)^2
    const float inv_2dx = 255.5f;      // (N-1)/2
    const float dt      = 1e-7f;

    const int  c    = base_c + (tid & 31);
    const int  r0   = base_r + (tid >> 5);
    const int  cc   = min(max(c, 1), GN - 2);
    const int  lc   = cc - base_c + 1;
    const float kap = (cc < GMB) ? k1 : k2;
    const float alp = (cc < GMB) ? a1 : a2;
    const bool iface = (cc == GMB - 1);

#pragma unroll
    for (int q = 0; q < 4; ++q) {
        const int r  = r0 + 8 * q;
        const int rr = min(max(r, 1), GN - 2);
        const int lr = rr - base_r + 1;

        const float Tc = s[lr * LDS_STRIDE + lc];
        const float Tm = s[(lr - 1) * LDS_STRIDE + lc];   // row rr-1 (axis 0)
        const float Tp = s[(lr + 1) * LDS_STRIDE + lc];   // row rr+1
        const float Tl = s[lr * LDS_STRIDE + lc - 1];     // col cc-1 (axis 1)
        const float Tr = s[lr * LDS_STRIDE + lc + 1];     // col cc+1

        float val;
        if (iface) {
            // interface: (k_left*T[:,ii+1] + k_right*T[:,ii-1]) / (kl+kr)
            val = (k1 * Tr + k2 * Tl) / (k1 + k2);
        } else {
            const float txx = (Tm - 2.0f * Tc + Tp) * inv_dx2;
            const float tyy = (Tl - 2.0f * Tc + Tr) * inv_dx2;
            const float tx_ = (Tp - Tm) * inv_2dx;
            const float ty_ = (Tr - Tl) * inv_2dx;
            const float t2  = Tc * Tc;
            const float t3  = t2 * Tc;
            val = Tc - dt * (t2 * tx_ - Tc * ty_) * kap
                     + dt * alp * (txx + tyy)
                     + dt * kap * (t3 - t2 + Tc);
        }
        Tout[r * GN + c] = val;
    }
}

extern "C" void kernel_launch(void* const* d_in, const int* in_sizes, int n_in,
                              void* d_out, int out_size, void* d_ws, size_t ws_size,
                              hipStream_t stream)
{
    (void)in_sizes; (void)n_in; (void)out_size; (void)d_ws; (void)ws_size;

    const float* u0 = (const float*)d_in[0];
    const float* k1 = (const float*)d_in[1];
    const float* k2 = (const float*)d_in[2];
    const float* a1 = (const float*)d_in[3];
    const float* a2 = (const float*)d_in[4];
    float* out = (float*)d_out;

    dim3 grid(GN / TILE, GN / TILE);   // 16 x 16 = 256 blocks
    dim3 block(NTHREADS);

    const float* src = u0;
    for (int t = 0; t < GSTEPS - 1; ++t) {
        float* dst = out + (size_t)t * GN * GN;
        adr2m_step<<<grid, block, 0, stream>>>(src, dst, k1, k2, a1, a2);
        src = dst;   // previous slab stays hot in 192MB L2
    }
}